// NodeMLP_76819785056585
// MI455X (gfx1250) — compile-verified
//
#include <hip/hip_runtime.h>
#include <hip/hip_bf16.h>
#include <math.h>

typedef __attribute__((ext_vector_type(2))) float v2f;
typedef __attribute__((ext_vector_type(8))) float v8f;

#define N_ROWS    4096
#define PRED_COLS 23
#define FDIM      32      // MLP output dim
#define CDIM      16      // conf dim
#define EPSV      1e-8f

// ---------------------------------------------------------------------------
// Stage 1: per-row MLP + normalize + sqrt(conf). 4096 rows, ~2.4K FMA each —
// negligible vs the 64MB output store of stage 2. One thread per row.
// ---------------------------------------------------------------------------
__global__ void prep_kernel(const float* __restrict__ preds,
                            const float* __restrict__ w1, const float* __restrict__ b1,
                            const float* __restrict__ w2, const float* __restrict__ b2,
                            float* __restrict__ A,   // [N_ROWS][FDIM] normalized features
                            float* __restrict__ S,   // [N_ROWS][CDIM] sqrt(conf)
                            float* __restrict__ C) { // [N_ROWS] class id
    int row = blockIdx.x * blockDim.x + threadIdx.x;
    if (row >= N_ROWS) return;
    const float* p = preds + row * PRED_COLS;
    float x0 = p[1], x1 = p[2], x2 = p[3], x3 = p[4], x4 = p[5];

    float h[64];
#pragma unroll
    for (int j = 0; j < 64; ++j) {
        const float* w = w1 + j * 5;
        float acc = fmaf(x0, w[0], b1[j]);
        acc = fmaf(x1, w[1], acc);
        acc = fmaf(x2, w[2], acc);
        acc = fmaf(x3, w[3], acc);
        acc = fmaf(x4, w[4], acc);
        h[j] = acc > 0.f ? acc : 0.f;
    }

    float f[FDIM];
    float nrm = 0.f;
#pragma unroll 4
    for (int k = 0; k < FDIM; ++k) {
        const float* w = w2 + k * 64;
        float acc = b2[k];
#pragma unroll
        for (int j = 0; j < 64; ++j) acc = fmaf(h[j], w[j], acc);
        f[k] = acc;
        nrm = fmaf(acc, acc, nrm);
    }
    float inv = 1.f / fmaxf(sqrtf(nrm), EPSV);
#pragma unroll
    for (int k = 0; k < FDIM; ++k) A[row * FDIM + k] = f[k] * inv;
#pragma unroll
    for (int c = 0; c < CDIM; ++c) S[row * CDIM + c] = sqrtf(p[7 + c]);
    C[row] = p[6];
}

// ---------------------------------------------------------------------------
// Stage 2: one wave per 16x16 output tile.
//   dist = n1 @ n2^T       (K=32 -> 8x V_WMMA_F32_16X16X4_F32)
//   conf = s1 @ s2^T       (K=16 -> 4x V_WMMA_F32_16X16X4_F32)
//   out  = (cls1==cls2) * conf * dist
// f32 WMMA A/B layout: lane<16 -> (idx=lane, K=0,1 in vgpr0,1);
//                      lane>=16 -> (idx=lane-16, K=2,3).
// C/D layout: vgpr v -> M = v + 8*(lane>=16), N = lane&15.
// ---------------------------------------------------------------------------
__global__ void __launch_bounds__(128)
pair_kernel(const float* __restrict__ A1, const float* __restrict__ S1,
            const float* __restrict__ C1,
            const float* __restrict__ A2, const float* __restrict__ S2,
            const float* __restrict__ C2,
            float* __restrict__ out) {
    const int lane  = threadIdx.x;                       // 0..31
    const int tileM = blockIdx.x * blockDim.y + threadIdx.y;  // 0..255
    const int tileN = blockIdx.y;                        // 0..255
    const int half  = lane >> 4;                         // 0 | 1
    const int l16   = lane & 15;
    const int koff  = half << 1;                         // K sub-offset 0 | 2

    const float* aRow  = A1 + (size_t)(tileM * 16 + l16) * FDIM + koff;
    const float* bRow  = A2 + (size_t)(tileN * 16 + l16) * FDIM + koff;
    const float* s1Row = S1 + (size_t)(tileM * 16 + l16) * CDIM + koff;
    const float* s2Row = S2 + (size_t)(tileN * 16 + l16) * CDIM + koff;

    v8f accD = {0.f, 0.f, 0.f, 0.f, 0.f, 0.f, 0.f, 0.f};
    v8f accC = {0.f, 0.f, 0.f, 0.f, 0.f, 0.f, 0.f, 0.f};

#pragma unroll
    for (int kc = 0; kc < FDIM; kc += 4) {
        v2f a = *(const v2f*)(aRow + kc);
        v2f b = *(const v2f*)(bRow + kc);
        accD = __builtin_amdgcn_wmma_f32_16x16x4_f32(false, a, false, b,
                                                     (short)0, accD, false, false);
    }
#pragma unroll
    for (int kc = 0; kc < CDIM; kc += 4) {
        v2f a = *(const v2f*)(s1Row + kc);
        v2f b = *(const v2f*)(s2Row + kc);
        accC = __builtin_amdgcn_wmma_f32_16x16x4_f32(false, a, false, b,
                                                     (short)0, accC, false, false);
    }

    const float c2    = C2[tileN * 16 + l16];            // this lane's column class
    const int   mBase = tileM * 16 + half * 8;
    float* outBase = out + (size_t)mBase * N_ROWS + (size_t)tileN * 16 + l16;

#pragma unroll
    for (int v = 0; v < 8; ++v) {
        float c1 = C1[mBase + v];
        float r  = (c1 == c2) ? accD[v] * accC[v] : 0.f;
        outBase[(size_t)v * N_ROWS] = r;
    }
}

// ---------------------------------------------------------------------------
extern "C" void kernel_launch(void* const* d_in, const int* in_sizes, int n_in,
                              void* d_out, int out_size, void* d_ws, size_t ws_size,
                              hipStream_t stream) {
    (void)in_sizes; (void)n_in; (void)out_size; (void)ws_size;
    const float* preds1 = (const float*)d_in[0];
    const float* preds2 = (const float*)d_in[1];
    const float* w1     = (const float*)d_in[2];
    const float* b1     = (const float*)d_in[3];
    const float* w2     = (const float*)d_in[4];
    const float* b2     = (const float*)d_in[5];
    float* out = (float*)d_out;

    float* ws = (float*)d_ws;   // 2 * (4096*32 + 4096*16 + 4096) floats ~= 1.6 MB
    float* A1 = ws;
    float* S1 = A1 + (size_t)N_ROWS * FDIM;
    float* C1 = S1 + (size_t)N_ROWS * CDIM;
    float* A2 = C1 + N_ROWS;
    float* S2 = A2 + (size_t)N_ROWS * FDIM;
    float* C2 = S2 + (size_t)N_ROWS * CDIM;

    prep_kernel<<<dim3(N_ROWS / 128), dim3(128), 0, stream>>>(preds1, w1, b1, w2, b2, A1, S1, C1);
    prep_kernel<<<dim3(N_ROWS / 128), dim3(128), 0, stream>>>(preds2, w1, b1, w2, b2, A2, S2, C2);

    // 256x256 tiles of 16x16; 4 waves (tiles) per block.
    pair_kernel<<<dim3(64, 256), dim3(32, 4), 0, stream>>>(A1, S1, C1, A2, S2, C2, out);
}